// LCF_Pooler_7610682048923
// MI455X (gfx1250) — compile-verified
//
#include <hip/hip_runtime.h>
#include <hip/hip_bf16.h>

typedef __attribute__((ext_vector_type(2))) float v2f;
typedef __attribute__((ext_vector_type(8))) float v8f;

#define B_ 32
#define S_ 2048
#define H_ 768

// ---------------------------------------------------------------------------
// Kernel 1: per-(b,s) row_sum = sum_h (lcf[b,s,h] - 1); mask = (row_sum == 0).
// One wave32 per row; coalesced lane-strided loads; shuffle-xor reduction.
// ---------------------------------------------------------------------------
__global__ void lcf_mask_kernel(const float* __restrict__ lcf,
                                int* __restrict__ mask) {
    const int row  = (blockIdx.x * blockDim.x + threadIdx.x) >> 5; // (b*S + s)
    const int lane = threadIdx.x & 31;
    if (row >= B_ * S_) return;
    const float* r = lcf + (size_t)row * H_;
    float s = 0.0f;
#pragma unroll 4
    for (int k = lane; k < H_; k += 32) s += r[k] - 1.0f;
    // wave32 reduction
#pragma unroll
    for (int off = 16; off > 0; off >>= 1) s += __shfl_xor(s, off, 32);
    if (lane == 0) mask[row] = (s == 0.0f) ? 1 : 0;
}

// ---------------------------------------------------------------------------
// Kernel 2: per-batch, find idx = position of the (count/2 + 1)-th masked s
// (argmax semantics: first match; if none, idx = 0), then gather that row of
// hidden_states into pooled[b][:].
// ---------------------------------------------------------------------------
__global__ void lcf_select_kernel(const int* __restrict__ mask,
                                  const float* __restrict__ hidden,
                                  float* __restrict__ pooled) {
    const int b   = blockIdx.x;
    const int tid = threadIdx.x;          // 0..255
    const int T   = 256;
    const int PER = S_ / T;               // 8 positions per thread

    __shared__ int cnt[256];
    __shared__ int pref[256];
    __shared__ int sh_target;
    __shared__ int sh_idx;

    const int* m = mask + b * S_;

    int c = 0;
#pragma unroll
    for (int i = 0; i < PER; ++i) c += m[tid * PER + i];
    cnt[tid] = c;
    if (tid == 0) sh_idx = 0x7FFFFFFF;
    __syncthreads();

    if (tid == 0) {
        int run = 0;
        for (int t = 0; t < T; ++t) { pref[t] = run; run += cnt[t]; }
        sh_target = run / 2 + 1;
    }
    __syncthreads();

    int run = pref[tid];
    const int target = sh_target;
#pragma unroll
    for (int i = 0; i < PER; ++i) {
        int mm = m[tid * PER + i];
        run += mm;
        if (mm && run == target) { atomicMin(&sh_idx, tid * PER + i); break; }
    }
    __syncthreads();

    int idx = sh_idx;
    if (idx >= S_) idx = 0;               // no masked position -> argmax = 0

    const float* src = hidden + ((size_t)b * S_ + idx) * H_;
    float* dst = pooled + b * H_;
    for (int h = tid; h < H_; h += T) dst[h] = src[h];
}

// ---------------------------------------------------------------------------
// Kernel 3: out = tanh(pooled @ W^T + bias) via V_WMMA_F32_16X16X4_F32.
// One wave per 16x16 output tile. grid = (H/16, B/16) = (48, 2).
//   A (16x4 f32):  lane<16 -> {K=k,k+1}, lane>=16 -> {K=k+2,k+3}, M = lane%16
//   B (4x16 f32):  same K split, N = lane%16; B[k][n] = W[n][k]
//   C/D (16x16):   VGPR r -> M = r + 8*(lane>=16), N = lane%16
// ---------------------------------------------------------------------------
__global__ void lcf_gemm_tanh_kernel(const float* __restrict__ pooled,
                                     const float* __restrict__ W,
                                     const float* __restrict__ bias,
                                     float* __restrict__ out) {
    const int nt   = blockIdx.x;          // N tile: 0..47
    const int mt   = blockIdx.y;          // M tile: 0..1
    const int lane = threadIdx.x;         // 0..31
    const int half = lane >> 4;
    const int l    = lane & 15;

    const float* arow = pooled + (size_t)(mt * 16 + l) * H_;  // A row (M=l)
    const float* brow = W      + (size_t)(nt * 16 + l) * H_;  // B col n=l -> W row n

    v8f c = {0.f, 0.f, 0.f, 0.f, 0.f, 0.f, 0.f, 0.f};

    for (int k0 = 0; k0 < H_; k0 += 4) {
        const int k = k0 + half * 2;
        v2f a, bb;
        a.x  = arow[k];
        a.y  = arow[k + 1];
        bb.x = brow[k];
        bb.y = brow[k + 1];
        c = __builtin_amdgcn_wmma_f32_16x16x4_f32(
                /*neg_a=*/false, a, /*neg_b=*/false, bb,
                /*c_mod=*/(short)0, c, /*reuse_a=*/false, /*reuse_b=*/false);
    }

#pragma unroll
    for (int r = 0; r < 8; ++r) {
        const int mm = mt * 16 + r + half * 8;
        const int nn = nt * 16 + l;
        out[mm * H_ + nn] = tanhf(c[r] + bias[nn]);
    }
}

// ---------------------------------------------------------------------------
// Launch
// ---------------------------------------------------------------------------
extern "C" void kernel_launch(void* const* d_in, const int* in_sizes, int n_in,
                              void* d_out, int out_size, void* d_ws, size_t ws_size,
                              hipStream_t stream) {
    const float* hidden = (const float*)d_in[0];   // (B, S, H) f32
    const float* lcf    = (const float*)d_in[1];   // (B, S, H) f32
    const float* W      = (const float*)d_in[2];   // (H, H)   f32
    const float* bias   = (const float*)d_in[3];   // (H,)     f32
    float* out          = (float*)d_out;           // (B, H)   f32

    int*   mask   = (int*)d_ws;                               // B*S ints (256 KB)
    float* pooled = (float*)((char*)d_ws + (size_t)B_ * S_ * sizeof(int)); // B*H f32

    // 1 wave per row, 8 waves per 256-thread block.
    const int rows   = B_ * S_;
    const int blocks = rows / 8;
    lcf_mask_kernel<<<blocks, 256, 0, stream>>>(lcf, mask);

    lcf_select_kernel<<<B_, 256, 0, stream>>>(mask, hidden, pooled);

    dim3 grid(H_ / 16, B_ / 16);
    lcf_gemm_tanh_kernel<<<grid, 32, 0, stream>>>(pooled, W, bias, out);
}